// GCNLinkPredictor_53558242181182
// MI455X (gfx1250) — compile-verified
//
#include <hip/hip_runtime.h>
#include <hip/hip_bf16.h>
#include <math.h>

// ---------------------------------------------------------------------------
// GCN link predictor for MI455X (gfx1250, wave32).
//   h = relu(gcnconv(x, W1, b1)); h = relu(gcnconv(h, W2, b2));
//   out = sigmoid([h[p0], h[p1]] @ fc_W + fc_b)
// GEMMs use V_WMMA_F32_16X16X4_F32 (exact fp32 path). W is staged in LDS
// pair-interleaved so each B operand is a single ds_load_b64 straight into an
// even VGPR pair (no repacking movs). Scatter-add uses native global fp32
// atomics; feature matrices (25.6 MB) are resident in the 192 MB L2, so the
// scatter is L2-bandwidth bound, not HBM bound.
// ---------------------------------------------------------------------------

typedef __attribute__((ext_vector_type(2))) float v2f;
typedef __attribute__((ext_vector_type(8))) float v8f;

#define CH 64

__device__ __forceinline__ void atomAddF(float* p, float v) {
  unsafeAtomicAdd(p, v);   // lowers to global_atomic_add_f32 on gfx1250
}

// ---- degree kernels -------------------------------------------------------
__global__ void k_deg_init(float* __restrict__ deg, int n) {
  int i = blockIdx.x * blockDim.x + threadIdx.x;
  if (i < n) deg[i] = 1.0f;                 // self loop contributes 1
}

__global__ void k_deg_accum(const int* __restrict__ dst, float* __restrict__ deg, int nE) {
  int e = blockIdx.x * blockDim.x + threadIdx.x;
  if (e < nE) atomAddF(&deg[dst[e]], 1.0f);
}

__global__ void k_rsqrt_inplace(float* __restrict__ d, int n) {
  int i = blockIdx.x * blockDim.x + threadIdx.x;
  if (i < n) d[i] = rsqrtf(d[i]);           // deg >= 1 always
}

// ---- fp32 WMMA GEMM: H[n,64] = X[n,64] @ W[64,64] -------------------------
// One wave computes a 16-row x 64-col strip via 4 accumulator tiles.
// A layout (16x4 f32): lanes 0-15 hold K=0,1 in v0,v1; lanes 16-31 hold K=2,3.
// B layout (4x16 f32): vgpr v, lane<16 -> (K=v, N=lane); lane>=16 -> (K=v+2, N=lane-16).
// LDS holds W pair-interleaved: Wp[p*64+c] = {W[2p][c], W[2p+1][c]}, so the
// per-lane B fragment for K-chunk kc is Wp[(2*kc+half)*64 + n*16 + l16], one b64.
__global__ void __launch_bounds__(256)
k_gemm64(const float* __restrict__ X, const float* __restrict__ W,
         float* __restrict__ H, int nRows, int nTiles) {
  __shared__ v2f Wp[32 * CH];               // 16 KB of the 320 KB WGP LDS
  for (int i = threadIdx.x; i < 32 * CH; i += 256) {
    const int p = i >> 6, c = i & 63;
    v2f w;
    w.x = W[(2 * p) * CH + c];
    w.y = W[(2 * p + 1) * CH + c];
    Wp[i] = w;
  }
  __syncthreads();

  const int wave = threadIdx.x >> 5;
  const int lane = threadIdx.x & 31;
  const int tile = blockIdx.x * 8 + wave;
  if (tile >= nTiles) return;               // wave-uniform exit: EXEC stays all-1s for WMMA

  const int row0 = tile * 16;
  const int half = lane >> 4;               // 0: K pair {0,1}; 1: K pair {2,3}
  const int l16  = lane & 15;
  const int arow = min(row0 + l16, nRows - 1);      // clamp (n is a multiple of 16 anyway)
  const float* xrow = X + (size_t)arow * CH + half * 2;

  v8f acc0 = {}, acc1 = {}, acc2 = {}, acc3 = {};
#pragma unroll
  for (int kc = 0; kc < 16; ++kc) {
    v2f a = *(const v2f*)(xrow + kc * 4);   // X[row][4kc + 2*half .. +1], 8B aligned
    const v2f* wp = &Wp[(2 * kc + half) * CH + l16];
    v2f b0 = wp[0];                         // cols n*16 .. : single ds_load_b64 each
    v2f b1 = wp[16];
    v2f b2 = wp[32];
    v2f b3 = wp[48];
    acc0 = __builtin_amdgcn_wmma_f32_16x16x4_f32(false, a, false, b0, (short)0, acc0, false, false);
    acc1 = __builtin_amdgcn_wmma_f32_16x16x4_f32(false, a, false, b1, (short)0, acc1, false, false);
    acc2 = __builtin_amdgcn_wmma_f32_16x16x4_f32(false, a, false, b2, (short)0, acc2, false, false);
    acc3 = __builtin_amdgcn_wmma_f32_16x16x4_f32(false, a, false, b3, (short)0, acc3, false, false);
  }

  // C/D layout: vgpr v, lane<16 -> (M=v, N=lane); lane>=16 -> (M=v+8, N=lane-16)
#pragma unroll
  for (int v = 0; v < 8; ++v) {
    const int r = row0 + v + half * 8;
    if (r < nRows) {
      float* hr = H + (size_t)r * CH + l16;
      hr[0]  = acc0[v];
      hr[16] = acc1[v];
      hr[32] = acc2[v];
      hr[48] = acc3[v];
    }
  }
}

// ---- scatter: agg[i] = h[i]*dinv[i]^2 (self loop term) --------------------
__global__ void k_selfloop_init(const float* __restrict__ h, const float* __restrict__ dinv,
                                float* __restrict__ agg, int n) {
  int t = blockIdx.x * blockDim.x + threadIdx.x;      // n*16 threads, float4 each
  if (t >= n * 16) return;
  int i = t >> 4, c = (t & 15) * 4;
  float w = dinv[i]; w *= w;
  float4 hv = *(const float4*)(h + (size_t)i * CH + c);
  float4 o = { hv.x * w, hv.y * w, hv.z * w, hv.w * w };
  *(float4*)(agg + (size_t)i * CH + c) = o;
}

// ---- scatter: agg[dst] += h[src] * dinv[src]*dinv[dst] over edges ---------
__global__ void k_edge_scatter(const int* __restrict__ src, const int* __restrict__ dst,
                               const float* __restrict__ dinv, const float* __restrict__ h,
                               float* __restrict__ agg, int nE) {
  long long t = (long long)blockIdx.x * blockDim.x + threadIdx.x;  // nE*16 threads
  if (t >= (long long)nE * 16) return;
  int e = (int)(t >> 4), c = ((int)t & 15) * 4;
  int s = src[e], d = dst[e];
  float norm = dinv[s] * dinv[d];
  float4 hv = *(const float4*)(h + (size_t)s * CH + c);  // 256B/edge, coalesced per 16 lanes
  float* ap = agg + (size_t)d * CH + c;
  atomAddF(ap + 0, hv.x * norm);
  atomAddF(ap + 1, hv.y * norm);
  atomAddF(ap + 2, hv.z * norm);
  atomAddF(ap + 3, hv.w * norm);
}

// ---- epilogue: agg = relu(agg + b), in place ------------------------------
__global__ void k_bias_relu(float* __restrict__ agg, const float* __restrict__ b, int n) {
  int t = blockIdx.x * blockDim.x + threadIdx.x;
  if (t >= n * 16) return;
  int i = t >> 4, c = (t & 15) * 4;
  float* p = agg + (size_t)i * CH + c;
  float4 v = *(const float4*)p;
  v.x = fmaxf(v.x + b[c + 0], 0.0f);
  v.y = fmaxf(v.y + b[c + 1], 0.0f);
  v.z = fmaxf(v.z + b[c + 2], 0.0f);
  v.w = fmaxf(v.w + b[c + 3], 0.0f);
  *(float4*)p = v;
}

// ---- head: sigmoid(concat(h[p0],h[p1]) @ fc_W + fc_b) ---------------------
__global__ void k_head(const float* __restrict__ act, const int* __restrict__ np,
                       const float* __restrict__ fcW, const float* __restrict__ fcb,
                       float* __restrict__ out) {
  __shared__ float red[128];
  int t = threadIdx.x;                       // 128 threads
  int node = np[t >> 6];                     // first 64: p0, next 64: p1
  red[t] = act[(size_t)node * CH + (t & 63)] * fcW[t];
  __syncthreads();
  for (int s = 64; s > 0; s >>= 1) {
    if (t < s) red[t] += red[t + s];
    __syncthreads();
  }
  if (t == 0) {
    float v = red[0] + fcb[0];
    out[0] = 1.0f / (1.0f + __expf(-v));
  }
}

// ---------------------------------------------------------------------------
extern "C" void kernel_launch(void* const* d_in, const int* in_sizes, int n_in,
                              void* d_out, int out_size, void* d_ws, size_t ws_size,
                              hipStream_t stream) {
  const float* x   = (const float*)d_in[0];
  const int*   ei  = (const int*)d_in[1];
  const int*   np  = (const int*)d_in[2];
  const float* W1  = (const float*)d_in[3];
  const float* b1  = (const float*)d_in[4];
  const float* W2  = (const float*)d_in[5];
  const float* b2  = (const float*)d_in[6];
  const float* fcW = (const float*)d_in[7];
  const float* fcb = (const float*)d_in[8];

  const int n  = in_sizes[0] / CH;          // 100000
  const int nE = in_sizes[1] / 2;           // 1600000
  const int* src = ei;
  const int* dst = ei + nE;

  // workspace layout: dinv | h | agg (all 512B aligned)
  char*  ws  = (char*)d_ws;
  size_t off = 0;
  auto carve = [&](size_t bytes) -> float* {
    float* p = (float*)(ws + off);
    off = (off + bytes + 511) & ~(size_t)511;
    return p;
  };
  float* dinv = carve((size_t)n * sizeof(float));
  float* h    = carve((size_t)n * CH * sizeof(float));
  float* agg  = carve((size_t)n * CH * sizeof(float));

  const int B = 256;
  const int nTiles     = (n + 15) / 16;
  const int gemmBlocks = (nTiles + 7) / 8;
  const int nodeBlocks = (int)(((long long)n * 16 + B - 1) / B);
  const int edgeBlocks = (int)(((long long)nE * 16 + B - 1) / B);

  // degrees (shared by both layers): dinv = rsqrt(1 + indeg)
  k_deg_init<<<(n + B - 1) / B, B, 0, stream>>>(dinv, n);
  k_deg_accum<<<(nE + B - 1) / B, B, 0, stream>>>(dst, dinv, nE);
  k_rsqrt_inplace<<<(n + B - 1) / B, B, 0, stream>>>(dinv, n);

  // layer 1
  k_gemm64<<<gemmBlocks, B, 0, stream>>>(x, W1, h, n, nTiles);
  k_selfloop_init<<<nodeBlocks, B, 0, stream>>>(h, dinv, agg, n);
  k_edge_scatter<<<edgeBlocks, B, 0, stream>>>(src, dst, dinv, h, agg, nE);
  k_bias_relu<<<nodeBlocks, B, 0, stream>>>(agg, b1, n);

  // layer 2 (agg holds act1; h is recycled for h2; then agg recycled for agg2)
  k_gemm64<<<gemmBlocks, B, 0, stream>>>(agg, W2, h, n, nTiles);
  k_selfloop_init<<<nodeBlocks, B, 0, stream>>>(h, dinv, agg, n);
  k_edge_scatter<<<edgeBlocks, B, 0, stream>>>(src, dst, dinv, h, agg, nE);
  k_bias_relu<<<nodeBlocks, B, 0, stream>>>(agg, b2, n);

  // head
  k_head<<<1, 128, 0, stream>>>(agg, np, fcW, fcb, (float*)d_out);
}